// GraphConvolution_90975997264005
// MI455X (gfx1250) — compile-verified
//
#include <hip/hip_runtime.h>
#include <stdint.h>

typedef __attribute__((ext_vector_type(2))) float v2f;
typedef __attribute__((ext_vector_type(8))) float v8f;

#define NROWS 16384
#define F 64
#define KB 64
#define LDA_S 68   // floats; row stride 272 B (16B-aligned), bank-conflict-free A reads
#define LDB_S 72   // floats; row stride 288 B (16B-aligned), half-wave groups disjoint banks

// ---------------- kernel 1: support = x @ W  (tiny, memory-trivial) ---------
__global__ __launch_bounds__(256) void gcn_xw(const float* __restrict__ x,
                                              const float* __restrict__ w,
                                              float* __restrict__ sup) {
  int idx = blockIdx.x * 256 + threadIdx.x;
  int i = idx >> 6;
  int j = idx & 63;
  const float* xr = x + (size_t)i * F;
  float acc = 0.f;
#pragma unroll
  for (int k = 0; k < F; ++k) acc = fmaf(xr[k], w[k * F + j], acc);
  sup[idx] = acc;
}

// Async global -> LDS copy of one 16-byte element (ASYNCcnt-tracked).
__device__ __forceinline__ void async_b128(uint32_t lds_addr, const float* gaddr) {
  asm volatile("global_load_async_to_lds_b128 %0, %1, off"
               :: "v"(lds_addr), "v"(gaddr)
               : "memory");
}

__device__ __forceinline__ void wait_async0() {
  asm volatile("s_wait_asynccnt 0x0" ::: "memory");
}

// Issue all async loads for one K-chunk into buffer pair (aBuf,bBuf).
__device__ __forceinline__ void issue_chunk(const float* __restrict__ adj_rows,
                                            const float* __restrict__ sup,
                                            int k0, float* aBuf, float* bBuf, int t) {
  // adj tile: 32 rows x 64 k = 512 float4 slots, 2 per thread
#pragma unroll
  for (int s = t; s < 512; s += 256) {
    int r  = s >> 4;
    int c4 = (s & 15) << 2;
    const float* gp = adj_rows + (size_t)r * NROWS + k0 + c4;
    async_b128((uint32_t)(uintptr_t)(aBuf + r * LDA_S + c4), gp);
  }
  // support tile: 64 k x 64 n = 1024 float4 slots, 4 per thread (L2-hot)
#pragma unroll
  for (int s = t; s < 1024; s += 256) {
    int r  = s >> 4;
    int c4 = (s & 15) << 2;
    const float* gp = sup + (size_t)(k0 + r) * F + c4;
    async_b128((uint32_t)(uintptr_t)(bBuf + r * LDB_S + c4), gp);
  }
}

// ---------------- kernel 2: out = adj @ support + bias  (WMMA f32) ----------
__global__ __launch_bounds__(256) void gcn_adj(const float* __restrict__ adj,
                                               const float* __restrict__ sup,
                                               const float* __restrict__ bias,
                                               float* __restrict__ out) {
  __shared__ __align__(16) float lA[2][32 * LDA_S];  // adj tiles (double-buffered)
  __shared__ __align__(16) float lB[2][KB * LDB_S];  // support tiles

  const int t    = threadIdx.x;
  const int lane = t & 31;
  const int wv   = t >> 5;     // wave 0..7
  const int rt   = wv >> 2;    // row-tile 0..1
  const int ct   = wv & 3;     // col-tile 0..3
  const int lm   = lane & 15;  // M / N position
  const int kh   = lane >> 4;  // half-wave selector (K offset 2*kh)

  const int row0 = blockIdx.x * 32;
  const int n0   = ct * 16;
  const float* adj_rows = adj + (size_t)row0 * NROWS;

  v8f acc = {};

  // Prologue: start streaming chunk 0 into buffer 0.
  issue_chunk(adj_rows, sup, 0, lA[0], lB[0], t);

  int p = 0;
  for (int k0 = 0; k0 < NROWS; k0 += KB) {
    wait_async0();        // this wave's issued copies have landed in LDS
    __syncthreads();      // all waves' copies landed; all reads of buf p^1 done

    if (k0 + KB < NROWS)  // uniform branch: stream next chunk while computing
      issue_chunk(adj_rows, sup, k0 + KB, lA[p ^ 1], lB[p ^ 1], t);

    const float* la = lA[p];
    const float* lb = lB[p];
#pragma unroll
    for (int kk = 0; kk < KB; kk += 4) {
      // A 16x4 f32: lane holds (M=lm, K=kk+2*kh+{0,1}) -> contiguous pair
      v2f a = *(const v2f*)&la[(rt * 16 + lm) * LDA_S + kk + 2 * kh];
      // B 4x16 f32: lane holds (K=kk+2*kh+{0,1}, N=n0+lm)
      v2f b;
      b.x = lb[(kk + 2 * kh) * LDB_S + n0 + lm];
      b.y = lb[(kk + 2 * kh + 1) * LDB_S + n0 + lm];
      acc = __builtin_amdgcn_wmma_f32_16x16x4_f32(false, a, false, b,
                                                  (short)0, acc, false, false);
    }
    p ^= 1;
  }

  // C/D layout: VGPR r, lane -> (M = r + 8*kh, N = lm)
  const int orow = row0 + rt * 16;
  const float bv = bias[n0 + lm];
#pragma unroll
  for (int r = 0; r < 8; ++r) {
    out[(size_t)(orow + r + 8 * kh) * F + n0 + lm] = acc[r] + bv;
  }
}

// ---------------- launcher --------------------------------------------------
extern "C" void kernel_launch(void* const* d_in, const int* in_sizes, int n_in,
                              void* d_out, int out_size, void* d_ws, size_t ws_size,
                              hipStream_t stream) {
  const float* x    = (const float*)d_in[0];
  const float* adj  = (const float*)d_in[1];
  const float* w    = (const float*)d_in[2];
  const float* bias = (const float*)d_in[3];
  float* out = (float*)d_out;
  float* sup = (float*)d_ws;  // 16384*64*4 = 4 MB scratch

  gcn_xw<<<(NROWS * F) / 256, 256, 0, stream>>>(x, w, sup);
  gcn_adj<<<NROWS / 32, 256, 0, stream>>>(adj, sup, bias, out);
}